// MeanAggregator_26963804685000
// MI455X (gfx1250) — compile-verified
//
#include <hip/hip_runtime.h>
#include <hip/hip_bf16.h>
#include <math.h>

// Shapes (fixed by the reference problem)
#define NNEI 50
#define DIM  128
#define ODIM 128
#define TOPK 16
#define INV_TEMP 2.0f   // 1 / 0.5
#define INV_N   0.02f   // 1 / 50

typedef float v2f __attribute__((ext_vector_type(2)));
typedef float v8f __attribute__((ext_vector_type(8)));
typedef unsigned int u32x4 __attribute__((ext_vector_type(4)));
typedef int i32x4 __attribute__((ext_vector_type(4)));
typedef int i32x8 __attribute__((ext_vector_type(8)));

#if __has_builtin(__builtin_amdgcn_tensor_load_to_lds)
  #define HAVE_TDM 1
  #if __has_include(<hip/amd_detail/amd_gfx1250_TDM.h>)
    #define TDM_6ARG 1   // amdgpu-toolchain (therock) 6-arg form
  #endif
#endif

// ---------------------------------------------------------------------------
// WMMA f32 16x16x4 helper: one K=4 step, D = A(16x4) * B(4x16) + C
// Lane mapping (ISA 7.12.2, wave32):
//   A: lanes 0-15 -> M=lane, VGPR0=K0 VGPR1=K1 ; lanes 16-31 -> M=lane-16, K2/K3
//   B: lanes 0-15 -> N=lane, VGPR0=K0 VGPR1=K1 ; lanes 16-31 -> N=lane-16, K2/K3
//   C/D: VGPR r -> lanes 0-15: (M=r, N=lane); lanes 16-31: (M=r+8, N=lane-16)
// ---------------------------------------------------------------------------
__device__ __forceinline__ v8f wmma_step_f32(v2f a, v2f b, v8f c) {
    return __builtin_amdgcn_wmma_f32_16x16x4_f32(
        /*neg_a=*/false, a, /*neg_b=*/false, b,
        /*c_mod=*/(short)0, c, /*reuse_a=*/false, /*reuse_b=*/false);
}

// ---------------------------------------------------------------------------
// Kernel 1: proj[B,128] = self_vecs[B,128] @ W_att[128,128]   (f32 WMMA)
// grid.x = B/16, block = 256 (8 waves); wave w owns output n-tile n0 = 16*w
// ---------------------------------------------------------------------------
__global__ void __launch_bounds__(256)
proj_gemm_kernel(const float* __restrict__ A,      // self_vecs [B,128]
                 const float* __restrict__ W,      // W_att [128,128]
                 float* __restrict__ out) {        // proj [B,128]
    const int wave = threadIdx.x >> 5;
    const int lane = threadIdx.x & 31;
    const int m0   = blockIdx.x * 16;
    const int n0   = wave * 16;
    const int rc   = lane & 15;      // row for A, col for B
    const int hi   = lane >> 4;      // K sub-pair selector

    v8f acc = {0.f, 0.f, 0.f, 0.f, 0.f, 0.f, 0.f, 0.f};
    const float* arow = A + (size_t)(m0 + rc) * DIM;
#pragma unroll 4
    for (int kk = 0; kk < DIM / 4; ++kk) {
        const int kbase = 4 * kk;
        v2f av = *(const v2f*)(arow + kbase + 2 * hi);            // K, K+1 contiguous
        v2f bv;
        bv[0] = W[(size_t)(kbase + 2 * hi + 0) * DIM + n0 + rc];
        bv[1] = W[(size_t)(kbase + 2 * hi + 1) * DIM + n0 + rc];
        acc = wmma_step_f32(av, bv, acc);
    }
#pragma unroll
    for (int r = 0; r < 8; ++r) {
        out[(size_t)(m0 + hi * 8 + r) * DIM + n0 + rc] = acc[r];
    }
}

// ---------------------------------------------------------------------------
// Kernel 2: per-row attention. One block (128 thr) per batch row.
// Single pass over neigh_vecs (the 419 MB tensor): TDM DMA (or async-to-LDS)
// stages the 50x128 tile in LDS once; reused for edge scores + weighted mean.
// ---------------------------------------------------------------------------
__global__ void __launch_bounds__(128)
attention_kernel(const float* __restrict__ neigh,   // [B,50,128]
                 const float* __restrict__ gumbel,  // [B,50]
                 const float* __restrict__ proj,    // [B,128]
                 float* __restrict__ nm,            // [B,128] workspace out
                 float* __restrict__ edge_out) {    // [B,50] output
    __shared__ float s_neigh[NNEI * DIM];   // 25.6 KB
    __shared__ float s_proj[DIM];
    __shared__ float s_edge[NNEI];
    __shared__ float s_val[NNEI];
    __shared__ float s_mask[NNEI];
    __shared__ float s_kth;

    const int b    = blockIdx.x;
    const int tid  = threadIdx.x;
    const int lane = tid & 31;
    const int wave = tid >> 5;

    const float* gsrc = neigh + (size_t)b * NNEI * DIM;

    // --- stage neigh row tile (50*128 f32 = 25.6 KB) into LDS ---------------
#if defined(HAVE_TDM)
    if (wave == 0) {   // TDM ignores EXEC: issue from exactly one wave
        const unsigned long long ga = (unsigned long long)(uintptr_t)gsrc;
        const unsigned int lds_off =
            (unsigned int)(unsigned long long)
            ( (__attribute__((address_space(3))) char*) s_neigh );

        // D# group 0 (ISA 8.3): count=1 | lds_addr | global_addr[56:0] | type=2
        u32x4 g0;
        g0[0] = 1u;                                           // count=1 (valid)
        g0[1] = lds_off;                                      // lds_addr (bytes)
        g0[2] = (unsigned int)(ga & 0xFFFFFFFFull);           // global_addr[31:0]
        g0[3] = (unsigned int)((ga >> 32) & 0x1FFFFFFull)     // global_addr[56:32]
                | (2u << 30);                                 // type=2 ("image")

        // D# group 1 (ISA 8.4), packed into 32-bit words:
        //   w0: wg_mask=0 | data_size=2(4B)<<16
        //   w1: atomic_barrier_addr=0 | tensor_dim0[15:0]<<16   (dim0=128)
        //   w2: tensor_dim0[31:16]=0  | tensor_dim1[15:0]<<16   (dim1=50)
        //   w3: tensor_dim1[31:16]=0  | tile_dim0<<16           (tile0=128)
        //   w4: tile_dim1=50 | tile_dim2=0
        //   w5: tensor_dim0_stride[31:0] = 128
        //   w6,w7: dim0_stride[47:32]=0, dim1_stride=0 (2D)
        i32x8 g1;
        g1[0] = (int)(2u << 16);
        g1[1] = (int)(128u << 16);
        g1[2] = (int)(50u << 16);
        g1[3] = (int)(128u << 16);
        g1[4] = 50;
        g1[5] = 128;
        g1[6] = 0;
        g1[7] = 0;
        i32x4 z4 = {0, 0, 0, 0};
#if defined(TDM_6ARG)
        i32x8 z8 = {0, 0, 0, 0, 0, 0, 0, 0};
        __builtin_amdgcn_tensor_load_to_lds(g0, g1, z4, z4, z8, 0);
#else
        __builtin_amdgcn_tensor_load_to_lds(g0, g1, z4, z4, 0);
#endif
        __builtin_amdgcn_s_wait_tensorcnt(0);
    }
    s_proj[tid] = proj[(size_t)b * DIM + tid];
#elif __has_builtin(__builtin_amdgcn_global_load_async_to_lds_b128)
    {
        const float4* g4 = (const float4*)gsrc;
        float4* s4 = (float4*)s_neigh;
        for (int i = tid; i < NNEI * DIM / 4; i += 128) {
            __builtin_amdgcn_global_load_async_to_lds_b128(
                (__attribute__((address_space(1))) void*)(g4 + i),
                (__attribute__((address_space(3))) void*)(s4 + i), 0, 0);
        }
        s_proj[tid] = proj[(size_t)b * DIM + tid];
        asm volatile("s_wait_asynccnt 0x0" ::: "memory");
    }
#else
    {
        const float4* g4 = (const float4*)gsrc;
        float4* s4 = (float4*)s_neigh;
        for (int i = tid; i < NNEI * DIM / 4; i += 128) s4[i] = g4[i];
        s_proj[tid] = proj[(size_t)b * DIM + tid];
    }
#endif
    __syncthreads();

    // --- edge_weight[n] = dot(neigh[n,:], proj[:]) ; wave w handles n=w,w+4,.. ---
    for (int n = wave; n < NNEI; n += 4) {
        float acc = 0.f;
#pragma unroll
        for (int i = 0; i < 4; ++i)
            acc += s_neigh[n * DIM + lane + 32 * i] * s_proj[lane + 32 * i];
#pragma unroll
        for (int off = 16; off >= 1; off >>= 1)
            acc += __shfl_xor(acc, off, 32);
        if (lane == 0) {
            s_edge[n] = acc;
            edge_out[(size_t)b * NNEI + n] = acc;   // raw scores are an output
        }
    }
    __syncthreads();

    // --- log_softmax + gumbel, scaled by 1/T ---
    float m = -INFINITY;
    for (int n = 0; n < NNEI; ++n) m = fmaxf(m, s_edge[n]);
    float S = 0.f;
    for (int n = 0; n < NNEI; ++n) S += __expf(s_edge[n] - m);
    const float logS = __logf(S);
    if (tid < NNEI) {
        float v = (s_edge[tid] - m - logS + gumbel[(size_t)b * NNEI + tid]) * INV_TEMP;
        s_val[tid] = v;
    }
    __syncthreads();

    // --- second softmax -> mask_values ---
    float m2 = -INFINITY;
    for (int n = 0; n < NNEI; ++n) m2 = fmaxf(m2, s_val[n]);
    float S2 = 0.f;
    for (int n = 0; n < NNEI; ++n) S2 += __expf(s_val[n] - m2);
    const float invS2 = 1.f / S2;
    if (tid < NNEI) s_mask[tid] = __expf(s_val[tid] - m2) * invS2;
    __syncthreads();

    // --- kth = min(top_k(mask_values, 16)) via stable rank (ties by index) ---
    if (tid < NNEI) {
        const float mv = s_mask[tid];
        int rank = 0;
        for (int j = 0; j < NNEI; ++j) {
            float oj = s_mask[j];
            rank += (oj > mv) || (oj == mv && j < tid);
        }
        if (rank == TOPK - 1) s_kth = mv;   // exactly one thread matches
    }
    __syncthreads();
    const float kth = s_kth;

    // --- weighted mean over neighbors (reuses LDS tile; ties included) ---
    float acc = 0.f;
    for (int n = 0; n < NNEI; ++n) {
        const float mv  = s_mask[n];
        const float sup = (mv >= kth) ? mv : 0.f;
        acc += sup * s_neigh[n * DIM + tid];
    }
    nm[(size_t)b * DIM + tid] = acc * INV_N;
}

// ---------------------------------------------------------------------------
// Kernel 3: out = relu(self @ self_w + nm @ neigh_w)   (f32 WMMA, fused K loop)
// grid.x = B/16, block = 256 (8 waves); wave w owns output n-tile n0 = 16*w
// ---------------------------------------------------------------------------
__global__ void __launch_bounds__(256)
output_gemm_kernel(const float* __restrict__ selfv,   // [B,128]
                   const float* __restrict__ nm,      // [B,128]
                   const float* __restrict__ self_w,  // [128,128]
                   const float* __restrict__ neigh_w, // [128,128]
                   float* __restrict__ out) {         // [B,128]
    const int wave = threadIdx.x >> 5;
    const int lane = threadIdx.x & 31;
    const int m0   = blockIdx.x * 16;
    const int n0   = wave * 16;
    const int rc   = lane & 15;
    const int hi   = lane >> 4;

    v8f acc = {0.f, 0.f, 0.f, 0.f, 0.f, 0.f, 0.f, 0.f};
#pragma unroll
    for (int pass = 0; pass < 2; ++pass) {
        const float* A = pass ? nm : selfv;
        const float* W = pass ? neigh_w : self_w;
        const float* arow = A + (size_t)(m0 + rc) * DIM;
#pragma unroll 4
        for (int kk = 0; kk < DIM / 4; ++kk) {
            const int kbase = 4 * kk;
            v2f av = *(const v2f*)(arow + kbase + 2 * hi);
            v2f bv;
            bv[0] = W[(size_t)(kbase + 2 * hi + 0) * ODIM + n0 + rc];
            bv[1] = W[(size_t)(kbase + 2 * hi + 1) * ODIM + n0 + rc];
            acc = wmma_step_f32(av, bv, acc);
        }
    }
#pragma unroll
    for (int r = 0; r < 8; ++r) {
        out[(size_t)(m0 + hi * 8 + r) * ODIM + n0 + rc] = fmaxf(acc[r], 0.f);
    }
}

// ---------------------------------------------------------------------------
extern "C" void kernel_launch(void* const* d_in, const int* in_sizes, int n_in,
                              void* d_out, int out_size, void* d_ws, size_t ws_size,
                              hipStream_t stream) {
    const float* self_vecs   = (const float*)d_in[0];   // [B,128]
    const float* neigh_vecs  = (const float*)d_in[1];   // [B,50,128]
    const float* gumbel      = (const float*)d_in[2];   // [B,50]
    const float* W_att       = (const float*)d_in[3];   // [128,128]
    const float* neigh_w     = (const float*)d_in[4];   // [128,128]
    const float* self_w      = (const float*)d_in[5];   // [128,128]

    const int B = in_sizes[0] / DIM;                    // 16384

    float* out_main = (float*)d_out;                    // [B,128]
    float* out_edge = out_main + (size_t)B * ODIM;      // [B,50]

    float* proj_ws = (float*)d_ws;                      // [B,128]
    float* nm_ws   = proj_ws + (size_t)B * DIM;         // [B,128]

    // 1) proj = self_vecs @ W_att            (WMMA f32)
    proj_gemm_kernel<<<B / 16, 256, 0, stream>>>(self_vecs, W_att, proj_ws);

    // 2) attention + gumbel-softmax top-k + weighted mean (TDM-staged LDS tile)
    attention_kernel<<<B, 128, 0, stream>>>(neigh_vecs, gumbel, proj_ws,
                                            nm_ws, out_edge);

    // 3) out = relu(self @ self_w + nm @ neigh_w)      (WMMA f32)
    output_gemm_kernel<<<B / 16, 256, 0, stream>>>(self_vecs, nm_ws,
                                                   self_w, neigh_w, out_main);
}